// TransformerBlock_71064528879765
// MI455X (gfx1250) — compile-verified
//
#include <hip/hip_runtime.h>

// ---- Problem constants (match reference) ----
#define D_MODEL   1024
#define NUM_HEADS 16
#define D_K       64
#define D_FF      2752
#define SEQ       2048
#define BATCH     2
#define ROWS      (BATCH*SEQ)   // 4096

typedef float v2f __attribute__((ext_vector_type(2)));
typedef float v8f __attribute__((ext_vector_type(8)));

// D = A(16x4) * B(4x16) + C, all fp32, full precision.
__device__ __forceinline__ v8f wmma_f32(v2f a, v2f b, v8f c) {
  return __builtin_amdgcn_wmma_f32_16x16x4_f32(
      /*neg_a=*/false, a, /*neg_b=*/false, b,
      /*c_mod=*/(short)0, c, /*reuse_a=*/false, /*reuse_b=*/false);
}

__device__ __forceinline__ float silu_mul(float u, float w) {
  return u * (1.0f / (1.0f + __expf(-u))) * w;
}

// =====================================================================
// RMSNorm: one block per row of D_MODEL
// =====================================================================
__global__ void __launch_bounds__(256)
rmsnorm_kernel(const float* __restrict__ x, const float* __restrict__ g,
               float* __restrict__ out, int d) {
  const int row = blockIdx.x;
  const float* xr = x + (size_t)row * d;
  float ss = 0.f;
  for (int i = threadIdx.x; i < d; i += blockDim.x) { float v = xr[i]; ss = fmaf(v, v, ss); }
#pragma unroll
  for (int m = 16; m >= 1; m >>= 1) ss += __shfl_xor(ss, m, 32);
  __shared__ float red[8];
  const int w = threadIdx.x >> 5, ln = threadIdx.x & 31;
  if (ln == 0) red[w] = ss;
  __syncthreads();
  float tot = 0.f;
#pragma unroll
  for (int i = 0; i < 8; i++) tot += red[i];
  const float inv = rsqrtf(tot / (float)d + 1e-5f);
  for (int i = threadIdx.x; i < d; i += blockDim.x)
    out[(size_t)row * d + i] = xr[i] * inv * g[i];
}

// =====================================================================
// RoPE (interleaved pairs within each 64-wide head), in place.
// one block per (b,s) row.
// =====================================================================
__global__ void __launch_bounds__(256)
rope_kernel(float* __restrict__ t, const int* __restrict__ positions,
            const int* __restrict__ theta_p) {
  const int row = blockIdx.x;           // b*SEQ + s
  const int s = row % SEQ;
  const float pos = (float)positions[s];
  const float th = (float)(*theta_p);
  float* r = t + (size_t)row * D_MODEL;
  for (int p = threadIdx.x; p < D_MODEL / 2; p += blockDim.x) {
    const int c0 = 2 * p;
    const int d  = c0 & (D_K - 1);      // even dim index within head
    const float freq = __powf(th, -(float)d * (1.0f / (float)D_K));
    const float ang = pos * freq;
    float sn, cs;
    sincosf(ang, &sn, &cs);
    const float xe = r[c0], xo = r[c0 + 1];
    r[c0]     = xe * cs - xo * sn;
    r[c0 + 1] = xe * sn + xo * cs;
  }
}

// =====================================================================
// WMMA GEMM: C[M,N] = A[M,K] @ B[N,K]^T  (both row-major along K)
// MODE 0: plain     MODE 1: + Res     MODE 2: A = silu(A)*A2, + Res
// One wave -> 32(M) x 64(N) strip = 8 accumulator tiles; K stepped by 4.
// Per k-step: 2 A-frag + 4 B-frag loads feed 8 WMMAs (0.75 loads/WMMA),
// and halving the M-tile count halves B re-reads from L2 vs a 16-row tile.
// =====================================================================
template <int MODE>
__global__ void __launch_bounds__(256)
gemm32(const float* __restrict__ A, const float* __restrict__ A2,
       const float* __restrict__ B, const float* __restrict__ Res,
       float* __restrict__ C, int M, int N, int K) {
  const int wid = blockIdx.x * 8 + (threadIdx.x >> 5);
  const int NT  = N >> 6;
  const int m0  = (wid / NT) << 5;      // 32-row tile
  const int n0  = (wid % NT) << 6;      // 64-col strip
  const int lane = threadIdx.x & 31;
  const int hf = lane >> 4;             // half-wave: K 0-1 vs 2-3 of the frag
  const int rc = lane & 15;             // A: row, B: col

  const float* Ar0 = A + (size_t)(m0 + rc) * K + 2 * hf;
  const float* Ar1 = Ar0 + (size_t)16 * K;
  const float* A2r0 = (MODE == 2) ? (A2 + (size_t)(m0 + rc) * K + 2 * hf) : nullptr;
  const float* A2r1 = (MODE == 2) ? (A2r0 + (size_t)16 * K) : nullptr;
  const float* B0  = B + (size_t)(n0 + rc) * K + 2 * hf;
  const size_t bstep = (size_t)16 * K;

  v8f acc[2][4] = {};
  for (int k0 = 0; k0 < K; k0 += 4) {
    v2f a0, a1;
    if (MODE == 2) {
      const v2f u0 = *(const v2f*)(Ar0 + k0);
      const v2f w0 = *(const v2f*)(A2r0 + k0);
      const v2f u1 = *(const v2f*)(Ar1 + k0);
      const v2f w1 = *(const v2f*)(A2r1 + k0);
      a0.x = silu_mul(u0.x, w0.x);  a0.y = silu_mul(u0.y, w0.y);
      a1.x = silu_mul(u1.x, w1.x);  a1.y = silu_mul(u1.y, w1.y);
    } else {
      a0 = *(const v2f*)(Ar0 + k0);
      a1 = *(const v2f*)(Ar1 + k0);
    }
    v2f b[4];
#pragma unroll
    for (int t = 0; t < 4; t++) b[t] = *(const v2f*)(B0 + (size_t)t * bstep + k0);
#pragma unroll
    for (int t = 0; t < 4; t++) {
      acc[0][t] = wmma_f32(a0, b[t], acc[0][t]);
      acc[1][t] = wmma_f32(a1, b[t], acc[1][t]);
    }
  }
#pragma unroll
  for (int i = 0; i < 2; i++) {
#pragma unroll
    for (int r = 0; r < 8; r++) {
      const size_t base = (size_t)(m0 + i * 16 + r + 8 * hf) * N + n0 + rc;
      float o0 = acc[i][0][r], o1 = acc[i][1][r], o2 = acc[i][2][r], o3 = acc[i][3][r];
      if (MODE >= 1) {
        o0 += Res[base];      o1 += Res[base + 16];
        o2 += Res[base + 32]; o3 += Res[base + 48];
      }
      C[base] = o0; C[base + 16] = o1; C[base + 32] = o2; C[base + 48] = o3;
    }
  }
}

// =====================================================================
// Flash attention (causal, online softmax), fp32 WMMA.
// One wave per 16-query tile of one (b,h). Scores never hit HBM
// (saves the 536 MB S matrix round-trip -> workload stays ~10us HBM-bound).
// Q/K/V/O layouts: [b, s, D_MODEL] with head offset h*D_K (row stride D_MODEL).
// =====================================================================
__global__ void __launch_bounds__(256)
flash_attn(const float* __restrict__ Q, const float* __restrict__ Kt,
           const float* __restrict__ V, float* __restrict__ O) {
  __shared__ float plds[8 * 256];                 // 16x16 P tile per wave
  const int wid = blockIdx.x * 8 + (threadIdx.x >> 5);
  const int NQT = SEQ / 16;
  const int qt = wid % NQT;
  const int bh = wid / NQT;
  const int h = bh % NUM_HEADS;
  const int b = bh / NUM_HEADS;
  const size_t base = (size_t)b * SEQ * D_MODEL + (size_t)h * D_K;
  const float* Qb = Q + base;
  const float* Kb = Kt + base;
  const float* Vb = V + base;
  float* Ob = O + base;
  const int lane = threadIdx.x & 31;
  const int hf = lane >> 4;
  const int rc = lane & 15;
  float* lp = plds + (threadIdx.x >> 5) * 256;

  // Resident Q A-fragments: 16 K-steps covering d_k = 64
  v2f qa[16];
#pragma unroll
  for (int i = 0; i < 16; i++)
    qa[i] = *(const v2f*)(Qb + (size_t)(qt * 16 + rc) * D_MODEL + i * 4 + 2 * hf);

  v8f o0 = {}, o1 = {}, o2 = {}, o3 = {};
  float mr[8], lr[8];
#pragma unroll
  for (int r = 0; r < 8; r++) { mr[r] = -1e30f; lr[r] = 0.f; }

  for (int jt = 0; jt <= qt; ++jt) {
    // S = Q K^T (16x16 tile), contraction over d_k
    v8f s = {};
    const float* Krow = Kb + (size_t)(jt * 16 + rc) * D_MODEL + 2 * hf;
#pragma unroll
    for (int i = 0; i < 16; i++) {
      const v2f kb = *(const v2f*)(Krow + i * 4);
      s = wmma_f32(qa[i], kb, s);
    }
    const bool diag = (jt == qt);
    // Online softmax per row; rows r+8*hf live on the 16 lanes of each half.
#pragma unroll
    for (int r = 0; r < 8; r++) {
      float sv = s[r] * 0.125f;                       // 1/sqrt(64)
      if (diag && (rc > r + 8 * hf)) sv = -1e30f;     // causal mask
      float mx = sv;
#pragma unroll
      for (int msk = 1; msk < 16; msk <<= 1) mx = fmaxf(mx, __shfl_xor(mx, msk, 32));
      const float mnew = fmaxf(mr[r], mx);
      const float corr = __expf(mr[r] - mnew);
      mr[r] = mnew;
      const float p = __expf(sv - mnew);
      float ps = p;
#pragma unroll
      for (int msk = 1; msk < 16; msk <<= 1) ps += __shfl_xor(ps, msk, 32);
      lr[r] = lr[r] * corr + ps;
      o0[r] *= corr; o1[r] *= corr; o2[r] *= corr; o3[r] *= corr;
      lp[(r + 8 * hf) * 16 + rc] = p;                 // stage P (D-layout -> LDS)
    }
    asm volatile("s_wait_dscnt 0" ::: "memory");      // same-wave LDS RAW fence
    // O += P V : contraction over 16 keys, 4 K-steps, 4 d_k tiles
#pragma unroll
    for (int kc = 0; kc < 4; kc++) {
      const v2f pa = *(const v2f*)(lp + rc * 16 + kc * 4 + 2 * hf);
      const float* Vr = Vb + (size_t)(jt * 16 + kc * 4 + 2 * hf) * D_MODEL;
      v2f vb0, vb1, vb2, vb3;
      vb0.x = Vr[rc];      vb0.y = Vr[D_MODEL + rc];
      vb1.x = Vr[16 + rc]; vb1.y = Vr[D_MODEL + 16 + rc];
      vb2.x = Vr[32 + rc]; vb2.y = Vr[D_MODEL + 32 + rc];
      vb3.x = Vr[48 + rc]; vb3.y = Vr[D_MODEL + 48 + rc];
      o0 = wmma_f32(pa, vb0, o0);
      o1 = wmma_f32(pa, vb1, o1);
      o2 = wmma_f32(pa, vb2, o2);
      o3 = wmma_f32(pa, vb3, o3);
    }
  }
#pragma unroll
  for (int r = 0; r < 8; r++) {
    const float inv = 1.0f / lr[r];
    const size_t ob = (size_t)(qt * 16 + r + 8 * hf) * D_MODEL + rc;
    Ob[ob]      = o0[r] * inv;
    Ob[ob + 16] = o1[r] * inv;
    Ob[ob + 32] = o2[r] * inv;
    Ob[ob + 48] = o3[r] * inv;
  }
}

// =====================================================================
// Orchestration. Workspace layout (floats):
//   xn | q | k | v | attn | x2 | x3      (x1 reuses the dead q/k/v region)
// Requires ws >= (6*ROWS*D_MODEL + ROWS*D_FF)*4 bytes ~= 146 MB.
// =====================================================================
extern "C" void kernel_launch(void* const* d_in, const int* in_sizes, int n_in,
                              void* d_out, int out_size, void* d_ws, size_t ws_size,
                              hipStream_t stream) {
  (void)in_sizes; (void)n_in; (void)out_size; (void)ws_size;
  const float* x        = (const float*)d_in[0];
  const int*   positions= (const int*)d_in[1];
  const int*   theta    = (const int*)d_in[2];
  const float* Wq = (const float*)d_in[3];
  const float* Wk = (const float*)d_in[4];
  const float* Wv = (const float*)d_in[5];
  const float* Wo = (const float*)d_in[6];
  const float* g1 = (const float*)d_in[7];
  const float* g2 = (const float*)d_in[8];
  const float* w1 = (const float*)d_in[9];
  const float* w2 = (const float*)d_in[10];
  const float* w3 = (const float*)d_in[11];
  float* out = (float*)d_out;

  float* xn   = (float*)d_ws;
  float* q    = xn   + (size_t)ROWS * D_MODEL;
  float* k    = q    + (size_t)ROWS * D_MODEL;
  float* v    = k    + (size_t)ROWS * D_MODEL;
  float* attn = v    + (size_t)ROWS * D_MODEL;
  float* x2   = attn + (size_t)ROWS * D_MODEL;
  float* x3b  = x2   + (size_t)ROWS * D_MODEL;
  float* x1b  = q;   // q/k/v dead after attention; 3*ROWS*D_MODEL > ROWS*D_FF

  const dim3 blk(256);
  const int gproj = (ROWS / 32) * (D_MODEL / 64) / 8;     // 256 blocks
  const int gff   = (ROWS / 32) * (D_FF / 64) / 8;        // 688 blocks
  const int gfa   = (BATCH * NUM_HEADS * (SEQ / 16)) / 8; // 512 blocks

  rmsnorm_kernel<<<ROWS, blk, 0, stream>>>(x, g1, xn, D_MODEL);
  gemm32<0><<<gproj, blk, 0, stream>>>(xn, nullptr, Wq, nullptr, q, ROWS, D_MODEL, D_MODEL);
  gemm32<0><<<gproj, blk, 0, stream>>>(xn, nullptr, Wk, nullptr, k, ROWS, D_MODEL, D_MODEL);
  gemm32<0><<<gproj, blk, 0, stream>>>(xn, nullptr, Wv, nullptr, v, ROWS, D_MODEL, D_MODEL);
  rope_kernel<<<ROWS, blk, 0, stream>>>(q, positions, theta);
  rope_kernel<<<ROWS, blk, 0, stream>>>(k, positions, theta);
  flash_attn<<<gfa, blk, 0, stream>>>(q, k, v, attn);
  gemm32<1><<<gproj, blk, 0, stream>>>(attn, nullptr, Wo, x, x2, ROWS, D_MODEL, D_MODEL);
  rmsnorm_kernel<<<ROWS, blk, 0, stream>>>(x2, g2, xn, D_MODEL);
  gemm32<0><<<gff, blk, 0, stream>>>(xn, nullptr, w1, nullptr, x1b, ROWS, D_FF, D_MODEL);
  gemm32<0><<<gff, blk, 0, stream>>>(xn, nullptr, w3, nullptr, x3b, ROWS, D_FF, D_MODEL);
  gemm32<2><<<gproj, blk, 0, stream>>>(x1b, x3b, w2, x2, out, ROWS, D_MODEL, D_FF);
}